// GLGModule_75093208203312
// MI455X (gfx1250) — compile-verified
//
#include <hip/hip_runtime.h>

typedef __attribute__((ext_vector_type(2))) float v2f;
typedef __attribute__((ext_vector_type(8))) float v8f;

// ---------------------------------------------------------------- zero fill
__global__ void k_zero(float4* __restrict__ p, size_t nv) {
  size_t i = (size_t)blockIdx.x * blockDim.x + threadIdx.x;
  if (i < nv) p[i] = make_float4(0.f, 0.f, 0.f, 0.f);
}

// ------------------------------------------------- edge scatter-add (segment_sum)
// 16 threads per edge, float4 per thread (64 features). Coalesced row gather,
// f32 atomics land in L2 (dest buffers << 192MB L2).
__global__ void k_agg(const float* __restrict__ x, const int* __restrict__ src,
                      const int* __restrict__ dst, float* __restrict__ out, int E) {
  long long t = (long long)blockIdx.x * blockDim.x + threadIdx.x;
  int e = (int)(t >> 4);
  if (e >= E) return;
  int fo = ((int)t & 15) << 2;
  const float4 v = *(const float4*)(x + (size_t)src[e] * 64 + fo);
  float* o = out + (size_t)dst[e] * 64 + fo;
  atomicAdd(o + 0, v.x);
  atomicAdd(o + 1, v.y);
  atomicAdd(o + 2, v.z);
  atomicAdd(o + 3, v.w);
}

// glg pass: source rows are the virtual concat [z4_g ; z4_l]
__global__ void k_agg_glg(const float* __restrict__ xg, const float* __restrict__ xl,
                          const int* __restrict__ src, const int* __restrict__ dst,
                          float* __restrict__ out, int E, int N) {
  long long t = (long long)blockIdx.x * blockDim.x + threadIdx.x;
  int e = (int)(t >> 4);
  if (e >= E) return;
  int fo = ((int)t & 15) << 2;
  int s = src[e];
  const float* base = (s < N) ? (xg + (size_t)s * 64) : (xl + (size_t)(s - N) * 64);
  const float4 v = *(const float4*)(base + fo);
  float* o = out + (size_t)dst[e] * 64 + fo;
  atomicAdd(o + 0, v.x);
  atomicAdd(o + 1, v.y);
  atomicAdd(o + 2, v.z);
  atomicAdd(o + 3, v.w);
}

// ----------------------------------------------- column sum over nodes (for mean)
__global__ void k_colsum(const float* __restrict__ z, int rows, float* __restrict__ gsum) {
  int t = blockIdx.x * blockDim.x + threadIdx.x;
  int f = t & 63;
  int r = t >> 6;
  int rstride = (gridDim.x * blockDim.x) >> 6;
  float s = 0.f;
  for (; r < rows; r += rstride) s += z[(size_t)r * 64 + f];
  atomicAdd(gsum + f, s);
}

// ---------------------------------- per-side constant vector: glob@W3 + sum_k b_k
__global__ void k_cvec(const float* __restrict__ gsum, const float* __restrict__ W,
                       const float* __restrict__ b, float inv_n, int K,
                       float* __restrict__ c) {
  int o = threadIdx.x;  // 64 threads
  float acc = 0.f;
  for (int k = 0; k < K; ++k) acc += b[k * 64 + o];
  const float* W3 = W + 3 * 64 * 64;
  float g = 0.f;
  for (int f = 0; f < 64; ++f) g += gsum[f] * W3[f * 64 + o];
  c[o] = acc + g * inv_n;
}

// --------------- combine + transpose weights: Wt[m][o][f]
// m=0: W0 + W_{K-1} (pairs with z4), m=1: W2 (deg*z4), m=2: W1 (y),
// m=3: W4 (hop after 1 agg),  m=4: W5 (hop after 2 aggs)
__global__ void k_wprep(const float* __restrict__ W, int K, float* __restrict__ Wt) {
  int t = blockIdx.x * blockDim.x + threadIdx.x;
  if (t >= 5 * 4096) return;
  int m = t >> 12;
  int o = (t >> 6) & 63;
  int f = t & 63;
  const int kidx[5] = {0, 2, 1, 4, 5};
  float v = W[((size_t)kidx[m] * 64 + f) * 64 + o];
  if (m == 0) v += W[((size_t)(K - 1) * 64 + f) * 64 + o];
  Wt[(size_t)m * 4096 + (size_t)o * 64 + f] = v;
}

// ---------------------------------------------------------------- dense stage
// One wave per 16-node tile. fp32 WMMA 16x16x4, K-stacked contraction:
//   out = z4@(W0+W6) + (deg*z4)@W2 + y@W1 + hop1@W4 + hop2@W5 + cvec
// A-frag: lane L holds A[L%16][k0 + 2*(L/16) + r], r=0,1  -> contiguous float2.
// B-frag: lane L holds B[k0 + 2*(L/16) + r][L%16]         -> contiguous float2
//         from Wt[m][o][k] (k-major).
// C/D:    VGPR j, lane L = (row j + 8*(L/16), col L%16).
// All source pointers are direct kernel args -> global_load (no FLAT).
// z4 fragment is loaded once and reused for both the plain and deg-scaled pass.
__global__ __launch_bounds__(256) void k_gemm(
    const float* __restrict__ zf, const float* __restrict__ h0,
    const float* __restrict__ h1, const float* __restrict__ y,
    const float* __restrict__ deg, const float* __restrict__ Wt,
    const float* __restrict__ cvec, float* __restrict__ out,
    int ntiles, int nrows) {
  int wid = threadIdx.x >> 5;
  int lane = threadIdx.x & 31;
  int tile = blockIdx.x * (blockDim.x >> 5) + wid;
  if (tile >= ntiles) return;                 // whole-wave uniform: EXEC stays all-1s
  int half = lane >> 4;
  int lm = lane & 15;
  int row = tile * 16 + lm;
  int rowc = row < nrows ? row : nrows - 1;   // clamp loads, keep EXEC full for WMMA
  float d = deg[rowc];

  v8f acc[4];
#pragma unroll
  for (int ot = 0; ot < 4; ++ot) {
    float cv = cvec[ot * 16 + lm];
    acc[ot] = (v8f){cv, cv, cv, cv, cv, cv, cv, cv};
  }

  const size_t roff = (size_t)rowc * 64 + 2 * half;
  const float* sz = zf + roff;
  const float* sy = y + roff;
  const float* s0 = h0 + roff;
  const float* s1 = h1 + roff;
  const float* wb = Wt + (size_t)lm * 64 + 2 * half;  // + m*4096 + ot*1024 + k0

  // z4 pass: feeds Wt[0] directly and Wt[1] after deg scaling (A loaded once)
#pragma unroll 4
  for (int k0 = 0; k0 < 64; k0 += 4) {
    v2f a = *(const v2f*)(sz + k0);
    v2f ad;
    ad.x = a.x * d;
    ad.y = a.y * d;
#pragma unroll
    for (int ot = 0; ot < 4; ++ot) {
      v2f b0 = *(const v2f*)(wb + ot * 1024 + k0);
      acc[ot] = __builtin_amdgcn_wmma_f32_16x16x4_f32(
          false, a, false, b0, (short)0, acc[ot], false, false);
      v2f b1 = *(const v2f*)(wb + 4096 + ot * 1024 + k0);
      acc[ot] = __builtin_amdgcn_wmma_f32_16x16x4_f32(
          false, ad, false, b1, (short)0, acc[ot], false, false);
    }
  }

  auto pass = [&](const float* s, const float* w) {
#pragma unroll 4
    for (int k0 = 0; k0 < 64; k0 += 4) {
      v2f a = *(const v2f*)(s + k0);
#pragma unroll
      for (int ot = 0; ot < 4; ++ot) {
        v2f b = *(const v2f*)(w + ot * 1024 + k0);
        acc[ot] = __builtin_amdgcn_wmma_f32_16x16x4_f32(
            false, a, false, b, (short)0, acc[ot], false, false);
      }
    }
  };
  pass(sy, wb + 2 * 4096);  // y    @ W1
  pass(s0, wb + 3 * 4096);  // hop1 @ W4
  pass(s1, wb + 4 * 4096);  // hop2 @ W5

#pragma unroll
  for (int ot = 0; ot < 4; ++ot) {
#pragma unroll
    for (int j = 0; j < 8; ++j) {
      int r = tile * 16 + j + 8 * half;
      if (r < nrows)
        __builtin_nontemporal_store(acc[ot][j], &out[(size_t)r * 64 + ot * 16 + lm]);
    }
  }
}

// ============================================================================
extern "C" void kernel_launch(void* const* d_in, const int* in_sizes, int n_in,
                              void* d_out, int out_size, void* d_ws, size_t ws_size,
                              hipStream_t stream) {
  const float* x_g     = (const float*)d_in[0];
  const float* x_lg    = (const float*)d_in[1];
  const float* deg_g   = (const float*)d_in[2];
  const float* deg_lg  = (const float*)d_in[3];
  const float* theta_W = (const float*)d_in[4];
  const float* theta_b = (const float*)d_in[5];
  const float* gamma_W = (const float*)d_in[6];
  const float* gamma_b = (const float*)d_in[7];
  const int* g_src   = (const int*)d_in[8];
  const int* g_dst   = (const int*)d_in[9];
  const int* lg_src  = (const int*)d_in[10];
  const int* lg_dst  = (const int*)d_in[11];
  const int* glg_src = (const int*)d_in[12];
  const int* glg_dst = (const int*)d_in[13];

  const int N = in_sizes[0] / 64;
  const int M = in_sizes[1] / 64;
  const int K = in_sizes[4] / 4096;   // 7 (radius 3 hop chain: 1,1,2 aggs)
  const int Eg = in_sizes[8], El = in_sizes[10], Eglg = in_sizes[12];

  float* ws = (float*)d_ws;
  size_t off = 0;
  auto take = [&](size_t n) { float* p = ws + off; off += n; return p; };
  float* h1g = take((size_t)N * 64);  // after 1 agg
  float* h2g = take((size_t)N * 64);  // after 2 aggs
  float* h4g = take((size_t)N * 64);  // after 4 aggs (= final = hop[2])
  float* tg  = take((size_t)N * 64);  // temp (after 3 aggs)
  float* h1l = take((size_t)M * 64);
  float* h2l = take((size_t)M * 64);
  float* h4l = take((size_t)M * 64);
  float* tl  = take((size_t)M * 64);
  float* yall   = take((size_t)(N + M) * 64);
  float* gsum_g = take(64);
  float* gsum_l = take(64);
  float* cvec_g = take(64);
  float* cvec_l = take(64);
  float* Wt_g = take(5 * 4096);
  float* Wt_l = take(5 * 4096);

  auto zero = [&](float* p, size_t n) {  // n in floats, multiple of 4
    size_t nv = n >> 2;
    int blocks = (int)((nv + 255) / 256);
    k_zero<<<blocks, 256, 0, stream>>>((float4*)p, nv);
  };
  auto agg = [&](const float* x, const int* s, const int* d, float* o, int E, int rows) {
    zero(o, (size_t)rows * 64);
    long long T = (long long)E * 16;
    int blocks = (int)((T + 255) / 256);
    k_agg<<<blocks, 256, 0, stream>>>(x, s, d, o, E);
  };

  // hop chains (radius 3): h1 -> h2 -> tmp -> h4
  agg(x_g, g_src, g_dst, h1g, Eg, N);
  agg(h1g, g_src, g_dst, h2g, Eg, N);
  agg(h2g, g_src, g_dst, tg,  Eg, N);
  agg(tg,  g_src, g_dst, h4g, Eg, N);

  agg(x_lg, lg_src, lg_dst, h1l, El, M);
  agg(h1l,  lg_src, lg_dst, h2l, El, M);
  agg(h2l,  lg_src, lg_dst, tl,  El, M);
  agg(tl,   lg_src, lg_dst, h4l, El, M);

  // glg mixed-graph pass on the concatenated final features
  zero(yall, (size_t)(N + M) * 64);
  {
    long long T = (long long)Eglg * 16;
    int blocks = (int)((T + 255) / 256);
    k_agg_glg<<<blocks, 256, 0, stream>>>(h4g, h4l, glg_src, glg_dst, yall, Eglg, N);
  }

  // global mean pool -> constant vectors
  zero(gsum_g, 128);  // gsum_g + gsum_l are contiguous
  k_colsum<<<256, 256, 0, stream>>>(h4g, N, gsum_g);
  k_colsum<<<256, 256, 0, stream>>>(h4l, M, gsum_l);
  k_cvec<<<1, 64, 0, stream>>>(gsum_g, theta_W, theta_b, 1.0f / (float)N, K, cvec_g);
  k_cvec<<<1, 64, 0, stream>>>(gsum_l, gamma_W, gamma_b, 1.0f / (float)M, K, cvec_l);

  // combined + transposed weights
  k_wprep<<<(5 * 4096 + 255) / 256, 256, 0, stream>>>(theta_W, K, Wt_g);
  k_wprep<<<(5 * 4096 + 255) / 256, 256, 0, stream>>>(gamma_W, K, Wt_l);

  // dense WMMA stage
  float* out = (float*)d_out;
  {
    int ntiles = (N + 15) / 16;
    int blocks = (ntiles + 7) / 8;
    k_gemm<<<blocks, 256, 0, stream>>>(h4g, h1g, h2g, yall, deg_g, Wt_g, cvec_g,
                                       out, ntiles, N);
  }
  {
    int ntiles = (M + 15) / 16;
    int blocks = (ntiles + 7) / 8;
    k_gemm<<<blocks, 256, 0, stream>>>(h4l, h1l, h2l, yall + (size_t)N * 64, deg_lg,
                                       Wt_l, cvec_l, out + (size_t)N * 64, ntiles, M);
  }
}